// HetSAGPooling_68753836474874
// MI455X (gfx1250) — compile-verified
//
#include <hip/hip_runtime.h>
#include <hip/hip_bf16.h>
#include <float.h>

#define D_DIM 256
#define B_SEG 64

typedef __attribute__((ext_vector_type(2))) float v2f;
typedef __attribute__((ext_vector_type(8))) float v8f;

__device__ __forceinline__ float gelu_exact(float x) {
    return 0.5f * x * (1.0f + erff(x * 0.70710678118654752f));
}

// ---------------- fill ----------------
__global__ void fill_kernel(float* __restrict__ p, float v, long n) {
    long i = (long)blockIdx.x * blockDim.x + threadIdx.x;
    long stride = (long)gridDim.x * blockDim.x;
    for (; i < n; i += stride) p[i] = v;
}

// ---------------- sum of a length-256 vector ----------------
__global__ void rowsum_kernel(const float* __restrict__ w, float* __restrict__ out) {
    __shared__ float s[D_DIM];
    s[threadIdx.x] = w[threadIdx.x];
    __syncthreads();
    for (int k = D_DIM / 2; k > 0; k >>= 1) {
        if ((int)threadIdx.x < k) s[threadIdx.x] += s[threadIdx.x + k];
        __syncthreads();
    }
    if (threadIdx.x == 0) out[0] = s[0];
}

// ---------------- kqv projection via V_WMMA_F32_16X16X4_F32 ----------------
// One wave computes a 16-node tile: C(16x16) = sum_t A(16x4 of x) * B(4x16 of Wkqv padded).
// A layout (32-bit 16x4): lanes 0-15 hold row m, K={4t,4t+1}; lanes 16-31 hold row m, K={4t+2,4t+3}.
// B layout (32-bit 4x16): vgpr0 lanes0-15 = K row 0, lanes16-31 = K row 2; vgpr1 = rows 1 / 3.
// C layout: vgpr r, lanes 0-15 -> [M=r][N=lane]; lanes 16-31 -> [M=8+r][N=lane-16].
__global__ __launch_bounds__(256) void kqv_wmma_kernel(
        const float* __restrict__ x,     // (n, 256)
        const float* __restrict__ W,     // (256, 3) row-major
        const float* __restrict__ bias,  // (3,)
        float* __restrict__ kv, float* __restrict__ qv, float* __restrict__ vv,
        int n) {
    const int lane = threadIdx.x & 31;
    const int wave = blockIdx.x * (blockDim.x >> 5) + (threadIdx.x >> 5);
    const int row0 = wave * 16;
    if (row0 >= n) return;                 // wave-uniform: EXEC stays all-ones for WMMA
    const int half = lane >> 4;            // selects K pair {0,1} vs {2,3}
    const int m = lane & 15;               // A row in tile / B,C column
    const int arow = min(row0 + m, n - 1);
    const float* xrow = x + (size_t)arow * D_DIM + 2 * half;

    v8f acc = {};
    for (int t = 0; t < D_DIM / 4; ++t) {
        v2f a = *(const v2f*)(xrow + 4 * t);
        float b0 = 0.f, b1 = 0.f;
        if (m < 3) {
            b0 = W[(4 * t + 2 * half + 0) * 3 + m];
            b1 = W[(4 * t + 2 * half + 1) * 3 + m];
        }
        v2f b; b.x = b0; b.y = b1;
        acc = __builtin_amdgcn_wmma_f32_16x16x4_f32(
            /*neg_a=*/false, a, /*neg_b=*/false, b,
            /*c_mod=*/(short)0, acc, /*reuse_a=*/false, /*reuse_b=*/false);
    }
    if (m < 3) {
        const float bb = bias[m];
        float* dst = (m == 0) ? kv : ((m == 1) ? qv : vv);
        #pragma unroll
        for (int r = 0; r < 8; ++r) {
            int rr = row0 + 8 * half + r;
            if (rr < n) dst[rr] = acc[r] + bb;
        }
    }
}

// ---------------- edge pass 1: segment max of logits ----------------
__global__ void edge_max_kernel(const int* __restrict__ ei, long E,
                                const float* __restrict__ qdst,
                                const float* __restrict__ ksrc,
                                const float* __restrict__ a_s,
                                const float* __restrict__ p_s,
                                float* __restrict__ mx) {
    const float scale = a_s[0] * p_s[0];   // / SQRT_HD == 1.0
    long i = (long)blockIdx.x * blockDim.x + threadIdx.x;
    long stride = (long)gridDim.x * blockDim.x;
    for (; i < E; i += stride) {
        int s = ei[i];
        int d = ei[E + i];
        float lg = qdst[d] * ksrc[s] * scale;
        atomicMax(mx + d, lg);             // global_atomic_max_num_f32
    }
}

// ---------------- edge pass 2: accumulate exp-sum and exp-weighted values ----------------
__global__ void edge_acc_kernel(const int* __restrict__ ei, long E,
                                const float* __restrict__ qdst,
                                const float* __restrict__ ksrc,
                                const float* __restrict__ vsrc,
                                const float* __restrict__ a_s,
                                const float* __restrict__ p_s,
                                const float* __restrict__ m_s,
                                const float* __restrict__ mx,
                                float* __restrict__ den,
                                float* __restrict__ num) {
    const float scale = a_s[0] * p_s[0];
    const float mval = m_s[0];
    long i = (long)blockIdx.x * blockDim.x + threadIdx.x;
    long stride = (long)gridDim.x * blockDim.x;
    for (; i < E; i += stride) {
        int s = ei[i];
        int d = ei[E + i];
        float lg = qdst[d] * ksrc[s] * scale;
        float e = __expf(lg - mx[d]);
        atomicAdd(den + d, e);
        atomicAdd(num + d, e * vsrc[s] * mval);
    }
}

// ---------------- per-node epilogue: agg -> gelu -> score ----------------
__global__ void score_kernel(const float* __restrict__ num,
                             const float* __restrict__ den,
                             const float* __restrict__ outw,
                             const float* __restrict__ outb,
                             const float* __restrict__ sumw,
                             const float* __restrict__ bsc,
                             float* __restrict__ score, int n) {
    long i = (long)blockIdx.x * blockDim.x + threadIdx.x;
    long stride = (long)gridDim.x * blockDim.x;
    for (; i < n; i += stride) {
        float agg = num[i] / (den[i] + 1e-16f);
        float attn = gelu_exact(agg) * outw[0] + outb[0];
        // (attn[:,None]*w + b).sum(-1) == attn*sum(w) + D*b
        score[i] = attn * sumw[0] + (float)D_DIM * bsc[0];
    }
}

// ---------------- batch segment max (B=64) with LDS staging ----------------
__global__ void batch_max_kernel(const float* __restrict__ score,
                                 const int* __restrict__ batch,
                                 float* __restrict__ bmax, int n) {
    __shared__ float lmax[B_SEG];
    if (threadIdx.x < B_SEG) lmax[threadIdx.x] = -FLT_MAX;
    __syncthreads();
    long i = (long)blockIdx.x * blockDim.x + threadIdx.x;
    long stride = (long)gridDim.x * blockDim.x;
    for (; i < n; i += stride) atomicMax(&lmax[batch[i]], score[i]);
    __syncthreads();
    if (threadIdx.x < B_SEG) atomicMax(&bmax[threadIdx.x], lmax[threadIdx.x]);
}

// ---------------- batch exp + segment sum; score[] becomes exp numerator ----------------
__global__ void batch_expsum_kernel(float* __restrict__ score,
                                    const int* __restrict__ batch,
                                    const float* __restrict__ bmax,
                                    float* __restrict__ bsum, int n) {
    __shared__ float ls[B_SEG];
    if (threadIdx.x < B_SEG) ls[threadIdx.x] = 0.f;
    __syncthreads();
    long i = (long)blockIdx.x * blockDim.x + threadIdx.x;
    long stride = (long)gridDim.x * blockDim.x;
    for (; i < n; i += stride) {
        int b = batch[i];
        float e = __expf(score[i] - bmax[b]);
        score[i] = e;
        atomicAdd(&ls[b], e);
    }
    __syncthreads();
    if (threadIdx.x < B_SEG) atomicAdd(&bsum[threadIdx.x], ls[threadIdx.x]);
}

// ---------------- pooled[b,:] += alpha_i * x[i,:] ; batch ids sorted ----------------
// Block handles 256 consecutive nodes, thread = feature dim; flush on segment change.
__global__ __launch_bounds__(256) void pool_kernel(const float* __restrict__ x,
                                                   const float* __restrict__ enum_,
                                                   const float* __restrict__ bsum,
                                                   const int* __restrict__ batch,
                                                   float* __restrict__ out, int n) {
    const int d = threadIdx.x;
    const int base = blockIdx.x * 256;
    float acc = 0.f;
    int curb = -1;
    for (int j = 0; j < 256; ++j) {
        int i = base + j;
        if (i >= n) break;
        int b = batch[i];
        if (b != curb) {
            if (curb >= 0) atomicAdd(out + (size_t)curb * D_DIM + d, acc);
            curb = b;
            acc = 0.f;
        }
        float w = enum_[i] / (bsum[b] + 1e-16f);
        acc = fmaf(w, x[(size_t)i * D_DIM + d], acc);
    }
    if (curb >= 0) atomicAdd(out + (size_t)curb * D_DIM + d, acc);
}

static inline int nblk(long n, int b) { return (int)((n + b - 1) / b); }

extern "C" void kernel_launch(void* const* d_in, const int* in_sizes, int n_in,
                              void* d_out, int out_size, void* d_ws, size_t ws_size,
                              hipStream_t stream) {
    const float* x_inst = (const float*)d_in[0];
    const float* x_net  = (const float*)d_in[1];
    const int*   ei_ii  = (const int*)d_in[2];
    const int*   ei_in  = (const int*)d_in[3];
    const int*   ei_ni  = (const int*)d_in[4];
    const int*   batch_inst = (const int*)d_in[5];
    const int*   batch_net  = (const int*)d_in[6];
    const float* Wkqv_inst = (const float*)d_in[7];
    const float* bkqv_inst = (const float*)d_in[8];
    const float* Wkqv_net  = (const float*)d_in[9];
    const float* bkqv_net  = (const float*)d_in[10];
    const float* a_ii = (const float*)d_in[11];
    const float* m_ii = (const float*)d_in[12];
    const float* p_ii = (const float*)d_in[13];
    const float* a_in = (const float*)d_in[14];
    const float* m_in = (const float*)d_in[15];
    const float* p_in = (const float*)d_in[16];
    const float* a_ni = (const float*)d_in[17];
    const float* m_ni = (const float*)d_in[18];
    const float* p_ni = (const float*)d_in[19];
    const float* outw_inst = (const float*)d_in[20];
    const float* outb_inst = (const float*)d_in[21];
    const float* outw_net  = (const float*)d_in[22];
    const float* outb_net  = (const float*)d_in[23];
    const float* w_inst = (const float*)d_in[24];
    const float* b_inst = (const float*)d_in[25];
    const float* w_net  = (const float*)d_in[26];
    const float* b_net  = (const float*)d_in[27];

    const int  Ni  = in_sizes[0] / D_DIM;
    const int  Nn  = in_sizes[1] / D_DIM;
    const long Eii = in_sizes[2] / 2;
    const long Ein = in_sizes[3] / 2;
    const long Eni = in_sizes[4] / 2;

    // workspace layout (floats)
    float* ws   = (float*)d_ws;
    float* k_i  = ws;            float* q_i = k_i + Ni;  float* v_i = q_i + Ni;
    float* k_n  = v_i + Ni;      float* q_n = k_n + Nn;  float* v_n = q_n + Nn;
    float* sc_i = v_n + Nn;      float* sc_n = sc_i + Ni;
    // zero-initialized region
    float* zero0  = sc_n + Nn;
    float* den_i  = zero0;            float* num_i  = den_i + Ni;
    float* den_n  = num_i + Ni;       float* num_n  = den_n + Nn;
    float* bsum_i = num_n + Nn;       float* bsum_n = bsum_i + B_SEG;
    const long zlen = 2L * Ni + 2L * Nn + 2L * B_SEG;
    // -FLT_MAX-initialized region
    float* neg0   = bsum_n + B_SEG;
    float* mx_i   = neg0;             float* mx_n   = mx_i + Ni;
    float* bmax_i = mx_n + Nn;        float* bmax_n = bmax_i + B_SEG;
    const long nlen = (long)Ni + Nn + 2L * B_SEG;
    float* sumw_i = bmax_n + B_SEG;   float* sumw_n = sumw_i + 1;

    // init
    fill_kernel<<<nblk(zlen, 256), 256, 0, stream>>>(zero0, 0.f, zlen);
    fill_kernel<<<nblk(out_size, 256), 256, 0, stream>>>((float*)d_out, 0.f, (long)out_size);
    fill_kernel<<<nblk(nlen, 256), 256, 0, stream>>>(neg0, -FLT_MAX, nlen);
    rowsum_kernel<<<1, 256, 0, stream>>>(w_inst, sumw_i);
    rowsum_kernel<<<1, 256, 0, stream>>>(w_net, sumw_n);

    // kqv projection (WMMA): 16 nodes per wave, 8 waves per block -> 128 nodes/block
    kqv_wmma_kernel<<<nblk(Ni, 128), 256, 0, stream>>>(x_inst, Wkqv_inst, bkqv_inst, k_i, q_i, v_i, Ni);
    kqv_wmma_kernel<<<nblk(Nn, 128), 256, 0, stream>>>(x_net,  Wkqv_net,  bkqv_net,  k_n, q_n, v_n, Nn);

    // segment softmax over edges: pass 1 (max)
    edge_max_kernel<<<nblk(Eii, 256), 256, 0, stream>>>(ei_ii, Eii, q_i, k_i, a_ii, p_ii, mx_i);
    edge_max_kernel<<<nblk(Eni, 256), 256, 0, stream>>>(ei_ni, Eni, q_i, k_n, a_ni, p_ni, mx_i);
    edge_max_kernel<<<nblk(Ein, 256), 256, 0, stream>>>(ei_in, Ein, q_n, k_i, a_in, p_in, mx_n);
    // pass 2 (exp-sum + exp-weighted value sum)
    edge_acc_kernel<<<nblk(Eii, 256), 256, 0, stream>>>(ei_ii, Eii, q_i, k_i, v_i, a_ii, p_ii, m_ii, mx_i, den_i, num_i);
    edge_acc_kernel<<<nblk(Eni, 256), 256, 0, stream>>>(ei_ni, Eni, q_i, k_n, v_n, a_ni, p_ni, m_ni, mx_i, den_i, num_i);
    edge_acc_kernel<<<nblk(Ein, 256), 256, 0, stream>>>(ei_in, Ein, q_n, k_i, v_i, a_in, p_in, m_in, mx_n, den_n, num_n);

    // node scores
    score_kernel<<<nblk(Ni, 256), 256, 0, stream>>>(num_i, den_i, outw_inst, outb_inst, sumw_i, b_inst, sc_i, Ni);
    score_kernel<<<nblk(Nn, 256), 256, 0, stream>>>(num_n, den_n, outw_net, outb_net, sumw_n, b_net, sc_n, Nn);

    // batch softmax (B=64)
    batch_max_kernel<<<nblk(Ni, 256), 256, 0, stream>>>(sc_i, batch_inst, bmax_i, Ni);
    batch_max_kernel<<<nblk(Nn, 256), 256, 0, stream>>>(sc_n, batch_net, bmax_n, Nn);
    batch_expsum_kernel<<<nblk(Ni, 256), 256, 0, stream>>>(sc_i, batch_inst, bmax_i, bsum_i, Ni);
    batch_expsum_kernel<<<nblk(Nn, 256), 256, 0, stream>>>(sc_n, batch_net, bmax_n, bsum_n, Nn);

    // weighted pooling into output
    float* out_inst = (float*)d_out;
    float* out_net  = out_inst + (size_t)B_SEG * D_DIM;
    pool_kernel<<<nblk(Ni, 256), 256, 0, stream>>>(x_inst, sc_i, bsum_i, batch_inst, out_inst, Ni);
    pool_kernel<<<nblk(Nn, 256), 256, 0, stream>>>(x_net,  sc_n, bsum_n, batch_net,  out_net, Nn);
}